// SPR_RGCN_88648124990576
// MI455X (gfx1250) — compile-verified
//
#include <hip/hip_runtime.h>

typedef __attribute__((ext_vector_type(2))) float v2f;
typedef __attribute__((ext_vector_type(8))) float v8f;

#define HIDF  64
#define NRELS 3
#define INDIM 80
#define NCLS  10
#define RT    2     // row tiles (of 16) per wave in the WMMA GEMM (kept small: no spills)

// ---------------- utility kernels ----------------
__global__ void zero_f32_kernel(float* p, int n) {
  int i = blockIdx.x * blockDim.x + threadIdx.x;
  int s = gridDim.x * blockDim.x;
  for (; i < n; i += s) p[i] = 0.0f;
}

__global__ void zero_i32_kernel(int* p, int n) {
  int i = blockIdx.x * blockDim.x + threadIdx.x;
  int s = gridDim.x * blockDim.x;
  for (; i < n; i += s) p[i] = 0;
}

// Build h0[N, 80] = concat(shape_emb[x0] (32), color_emb[x1] (32), pos_emb[x2] (16))
__global__ void embed_kernel(const float* __restrict__ shape_emb,
                             const float* __restrict__ color_emb,
                             const float* __restrict__ pos_emb,
                             const int* __restrict__ x,
                             float* __restrict__ h, int N) {
  int i = blockIdx.x * blockDim.x + threadIdx.x;
  int s = gridDim.x * blockDim.x;
  int total = N * INDIM;
  for (; i < total; i += s) {
    int n = i / INDIM;
    int f = i - n * INDIM;
    float v;
    if (f < 32)       v = shape_emb[x[n * 3 + 0] * 32 + f];
    else if (f < 64)  v = color_emb[x[n * 3 + 1] * 32 + (f - 32)];
    else              v = pos_emb[x[n * 3 + 2] * 16 + (f - 64)];
    h[i] = v;
  }
}

// cnt[d][r] = number of incoming edges at node d with relation r
__global__ void count_kernel(const int* __restrict__ dst,
                             const int* __restrict__ et,
                             int* __restrict__ cnt, int E) {
  int i = blockIdx.x * blockDim.x + threadIdx.x;
  int s = gridDim.x * blockDim.x;
  for (; i < E; i += s)
    atomicAdd(cnt + (size_t)dst[i] * NRELS + et[i], 1);
}

// ---------------- WMMA GEMM: out[N x 64] = A[N x K] @ W[K x 64] (+ bias) ----------------
// One wave per RT*16-row macro tile; each B fragment is reused across RT row tiles
// (RT*4 WMMAs per K-step for 8 B-loads + RT A-loads).
// A fragment (16x4 f32): row = lane&15; lanes<16 hold K {0,1}, lanes>=16 hold K {2,3}.
// B fragment (4x16 f32): same K split; col = lane&15.
__global__ void __launch_bounds__(256, 1)
gemm_wmma_kernel(const float* __restrict__ A,
                 const float* __restrict__ W,
                 const float* __restrict__ bias,
                 float* __restrict__ out,
                 int n_rows, int K) {
  int wave = blockIdx.x * (blockDim.x >> 5) + (threadIdx.x >> 5);
  int lane = threadIdx.x & 31;
  int base = wave * (16 * RT);
  if (base >= n_rows) return;            // uniform per wave -> EXEC stays all-ones

  int m  = lane & 15;
  int kh = (lane >> 4) << 1;             // 0 for lanes 0-15, 2 for lanes 16-31

  int trow[RT];
  const float* arow[RT];
#pragma unroll
  for (int t = 0; t < RT; ++t) {
    int r0 = base + 16 * t;
    if (r0 > n_rows - 16) r0 = n_rows - 16;  // clamp (n_rows is a multiple of 16);
    trow[t] = r0;                            // duplicated tiles store identical data
    arow[t] = A + (size_t)(r0 + m) * K + kh;
  }

  v8f acc[RT][4];
#pragma unroll
  for (int t = 0; t < RT; ++t)
#pragma unroll
    for (int c = 0; c < 4; ++c) acc[t][c] = (v8f){};

  for (int k = 0; k < K; k += 4) {
    const float* w0 = W + (size_t)(k + kh) * HIDF;
    v2f b[4];
#pragma unroll
    for (int c = 0; c < 4; ++c) {
      b[c].x = w0[c * 16 + m];
      b[c].y = w0[HIDF + c * 16 + m];
    }
#pragma unroll
    for (int t = 0; t < RT; ++t) {
      v2f a;
      a.x = arow[t][k];
      a.y = arow[t][k + 1];
#pragma unroll
      for (int c = 0; c < 4; ++c)
        acc[t][c] = __builtin_amdgcn_wmma_f32_16x16x4_f32(
            false, a, false, b[c], (short)0, acc[t][c], false, false);
    }
  }

  float bv[4] = {0.f, 0.f, 0.f, 0.f};
  if (bias) {
#pragma unroll
    for (int c = 0; c < 4; ++c) bv[c] = bias[c * 16 + m];
  }

  // C/D layout: VGPR i -> row (i + 8*(lane>=16)), col = (lane&15) + 16*c
  int roff = (lane >> 4) << 3;
#pragma unroll
  for (int t = 0; t < RT; ++t) {
#pragma unroll
    for (int i = 0; i < 8; ++i) {
      float* o = out + (size_t)(trow[t] + roff + i) * HIDF + m;
#pragma unroll
      for (int c = 0; c < 4; ++c) o[c * 16] = acc[t][c][i] + bv[c];
    }
  }
}

// ---------------- edge scatter: acc[dst] += t[rel][src] / max(cnt[dst][rel],1) ----------------
// 16 threads per edge, 4 floats each (float4 gather, scalar f32 atomics for scatter)
__global__ void scatter_kernel(const float* __restrict__ t,
                               const int* __restrict__ src,
                               const int* __restrict__ dst,
                               const int* __restrict__ et,
                               const int* __restrict__ cnt,
                               float* __restrict__ acc, int E, int N) {
  int i = blockIdx.x * blockDim.x + threadIdx.x;
  int s = gridDim.x * blockDim.x;
  int total = E * 16;
  for (; i < total; i += s) {
    int e  = i >> 4;
    int fq = (i & 15) << 2;
    int r  = et[e];
    int sn = src[e];
    int dn = dst[e];
    int c  = cnt[(size_t)dn * NRELS + r];
    float scale = 1.0f / (float)(c > 0 ? c : 1);
    const float4 v = *(const float4*)(t + ((size_t)r * N + sn) * HIDF + fq);
    float* ap = acc + (size_t)dn * HIDF + fq;
    atomicAdd(ap + 0, v.x * scale);
    atomicAdd(ap + 1, v.y * scale);
    atomicAdd(ap + 2, v.z * scale);
    atomicAdd(ap + 3, v.w * scale);
  }
}

__global__ void relu_kernel(float* __restrict__ p, int n) {
  int i = blockIdx.x * blockDim.x + threadIdx.x;
  int s = gridDim.x * blockDim.x;
  for (; i < n; i += s) p[i] = fmaxf(p[i], 0.0f);
}

// global mean pool: gsum[g][f] += h[n][f], gcnt[g] += 1
__global__ void pool_kernel(const float* __restrict__ h,
                            const int* __restrict__ batch,
                            float* __restrict__ gsum,
                            int* __restrict__ gcnt, int N) {
  int i = blockIdx.x * blockDim.x + threadIdx.x;
  int s = gridDim.x * blockDim.x;
  int total = N * HIDF;
  for (; i < total; i += s) {
    int n = i / HIDF;
    int f = i - n * HIDF;
    int g = batch[n];
    atomicAdd(gsum + (size_t)g * HIDF + f, h[i]);
    if (f == 0) atomicAdd(gcnt + g, 1);
  }
}

// out[g][c] = (gsum[g]/max(gcnt,1)) . linW[:,c] + linb[c]
__global__ void final_kernel(const float* __restrict__ gsum,
                             const int* __restrict__ gcnt,
                             const float* __restrict__ linW,
                             const float* __restrict__ linb,
                             float* __restrict__ out, int G) {
  int i = blockIdx.x * blockDim.x + threadIdx.x;
  int s = gridDim.x * blockDim.x;
  int total = G * NCLS;
  for (; i < total; i += s) {
    int g = i / NCLS;
    int c = i - g * NCLS;
    int cn = gcnt[g];
    float inv = 1.0f / (float)(cn > 0 ? cn : 1);
    float acc = linb[c];
    const float* gs = gsum + (size_t)g * HIDF;
    for (int k = 0; k < HIDF; ++k)
      acc += gs[k] * inv * linW[k * NCLS + c];
    out[i] = acc;
  }
}

// ---------------- host-side launch ----------------
extern "C" void kernel_launch(void* const* d_in, const int* in_sizes, int n_in,
                              void* d_out, int out_size, void* d_ws, size_t ws_size,
                              hipStream_t stream) {
  const float* shape_emb = (const float*)d_in[0];
  const float* color_emb = (const float*)d_in[1];
  const float* pos_emb   = (const float*)d_in[2];
  const float* W1        = (const float*)d_in[3];
  const float* root1     = (const float*)d_in[4];
  const float* b1        = (const float*)d_in[5];
  const float* W2        = (const float*)d_in[6];
  const float* root2     = (const float*)d_in[7];
  const float* b2        = (const float*)d_in[8];
  const float* linW      = (const float*)d_in[9];
  const float* linb      = (const float*)d_in[10];
  const int*   x         = (const int*)d_in[11];
  const int*   edge_idx  = (const int*)d_in[12];
  const int*   edge_type = (const int*)d_in[13];
  const int*   batch     = (const int*)d_in[14];

  const int N = in_sizes[14];
  const int E = in_sizes[13];
  const int G = out_size / NCLS;
  const int* src = edge_idx;
  const int* dst = edge_idx + E;

  // workspace layout (all 16B-aligned given 256B-aligned d_ws)
  float* h0   = (float*)d_ws;                         // N * 80 (layer-1 feats; reused as layer-2 acc)
  float* acc  = h0  + (size_t)N * INDIM;              // N * 64 (layer-1 acc -> h1)
  float* t    = acc + (size_t)N * HIDF;               // 3 * N * 64 (per-relation transformed feats)
  int*   cnt  = (int*)(t + (size_t)NRELS * N * HIDF); // N * 3
  float* gsum = (float*)(cnt + (size_t)N * NRELS);    // G * 64
  int*   gcnt = (int*)(gsum + (size_t)G * HIDF);      // G

  const int B = 256;
  auto grid = [](long n, int b) { return (int)((n + b - 1) / b); };

  // zero counters / pool accumulators
  zero_i32_kernel<<<grid((long)N * NRELS, B), B, 0, stream>>>(cnt, N * NRELS);
  zero_f32_kernel<<<grid((long)G * HIDF, B), B, 0, stream>>>(gsum, G * HIDF);
  zero_i32_kernel<<<grid(G, B), B, 0, stream>>>(gcnt, G);

  // node features + edge counts
  embed_kernel<<<grid((long)N * INDIM, B), B, 0, stream>>>(shape_emb, color_emb, pos_emb, x, h0, N);
  count_kernel<<<grid(E, B), B, 0, stream>>>(dst, edge_type, cnt, E);

  const int waves   = (N + 16 * RT - 1) / (16 * RT);
  const int gBlocks = (waves + 7) / 8;   // 8 waves per 256-thread block

  // ---- layer 1: acc = h0@root1 + b1 ; t_r = h0@W1_r ; scatter ; relu ----
  gemm_wmma_kernel<<<gBlocks, B, 0, stream>>>(h0, root1, b1, acc, N, INDIM);
  for (int r = 0; r < NRELS; ++r)
    gemm_wmma_kernel<<<gBlocks, B, 0, stream>>>(h0, W1 + (size_t)r * INDIM * HIDF, nullptr,
                                                t + (size_t)r * N * HIDF, N, INDIM);
  scatter_kernel<<<grid((long)E * 16, B), B, 0, stream>>>(t, src, dst, edge_type, cnt, acc, E, N);
  relu_kernel<<<grid((long)N * HIDF, B), B, 0, stream>>>(acc, N * HIDF);

  // ---- layer 2: h0buf = acc@root2 + b2 ; t_r = acc@W2_r ; scatter ; relu ----
  gemm_wmma_kernel<<<gBlocks, B, 0, stream>>>(acc, root2, b2, h0, N, HIDF);
  for (int r = 0; r < NRELS; ++r)
    gemm_wmma_kernel<<<gBlocks, B, 0, stream>>>(acc, W2 + (size_t)r * HIDF * HIDF, nullptr,
                                                t + (size_t)r * N * HIDF, N, HIDF);
  scatter_kernel<<<grid((long)E * 16, B), B, 0, stream>>>(t, src, dst, edge_type, cnt, h0, E, N);
  relu_kernel<<<grid((long)N * HIDF, B), B, 0, stream>>>(h0, N * HIDF);

  // ---- pool + classifier ----
  pool_kernel<<<grid((long)N * HIDF, B), B, 0, stream>>>(h0, batch, gsum, gcnt, N);
  final_kernel<<<grid((long)G * NCLS, B), B, 0, stream>>>(gsum, gcnt, linW, linb, (float*)d_out, G);
}